// Mae_74088185856822
// MI455X (gfx1250) — compile-verified
//
#include <hip/hip_runtime.h>
#include <cstdint>

#define HD __host__ __device__ __forceinline__

HD uint32_t rotl32(uint32_t v, int r) { return (v << r) | (v >> (32 - r)); }

// JAX Threefry-2x32, 20 rounds (rotation sets {13,15,26,6} / {17,29,16,24}).
HD void threefry2x32(uint32_t k0, uint32_t k1, uint32_t x0, uint32_t x1,
                     uint32_t& o0, uint32_t& o1) {
  const uint32_t k2 = 0x1BD11BDAu ^ k0 ^ k1;
  x0 += k0; x1 += k1;
#define TF_R(r) { x0 += x1; x1 = rotl32(x1, (r)); x1 ^= x0; }
  TF_R(13) TF_R(15) TF_R(26) TF_R(6)   x0 += k1; x1 += k2 + 1u;
  TF_R(17) TF_R(29) TF_R(16) TF_R(24)  x0 += k2; x1 += k0 + 2u;
  TF_R(13) TF_R(15) TF_R(26) TF_R(6)   x0 += k0; x1 += k1 + 3u;
  TF_R(17) TF_R(29) TF_R(16) TF_R(24)  x0 += k1; x1 += k2 + 4u;
  TF_R(13) TF_R(15) TF_R(26) TF_R(6)   x0 += k2; x1 += k0 + 5u;
#undef TF_R
  o0 = x0; o1 = x1;
}

// JAX uniform [0,1): (bits>>9)|0x3f800000 bitcast - 1.0
__device__ __forceinline__ float u01(uint32_t b) {
  return __uint_as_float((b >> 9) | 0x3f800000u) - 1.0f;
}

// JAX normal: sqrt(2) * erfinv(uniform(nextafter(-1,0), 1)) -- XLA ErfInv f32 poly.
__device__ __forceinline__ float jax_normal(uint32_t bits) {
  const float lo = -0.99999994f;  // nextafter(-1, 0)
  float u = u01(bits) * (1.0f - lo) + lo;
  u = fmaxf(lo, u);
  float w = -log1pf(-u * u);
  float p;
  if (w < 5.0f) {
    w -= 2.5f;
    p = 2.81022636e-08f;
    p = fmaf(p, w, 3.43273939e-07f);
    p = fmaf(p, w, -3.5233877e-06f);
    p = fmaf(p, w, -4.39150654e-06f);
    p = fmaf(p, w, 0.00021858087f);
    p = fmaf(p, w, -0.00125372503f);
    p = fmaf(p, w, -0.00417768164f);
    p = fmaf(p, w, 0.246640727f);
    p = fmaf(p, w, 1.50140941f);
  } else {
    w = sqrtf(w) - 3.0f;
    p = -0.000200214257f;
    p = fmaf(p, w, 0.000100950558f);
    p = fmaf(p, w, 0.00134934322f);
    p = fmaf(p, w, -0.00367342844f);
    p = fmaf(p, w, 0.00573950773f);
    p = fmaf(p, w, -0.0076224613f);
    p = fmaf(p, w, 0.00943887047f);
    p = fmaf(p, w, 1.00167406f);
    p = fmaf(p, w, 2.83297682f);
  }
  return 1.41421356f * (p * u);
}

// Kernel A: per-patch Bernoulli(0.7) mask, 65536 patches, JAX pairing (i, i+32768).
__global__ __launch_bounds__(256) void mae_mask_kernel(unsigned char* __restrict__ mask,
                                                       uint32_t km0, uint32_t km1) {
  const uint32_t i = blockIdx.x * 256u + threadIdx.x;  // 32768 threads
  uint32_t b0, b1;
  threefry2x32(km0, km1, i, i + 32768u, b0, b1);
  mask[i]           = (u01(b0) < 0.7f) ? 1 : 0;
  mask[i + 32768u]  = (u01(b1) < 0.7f) ? 1 : 0;
}

// Kernel B: each thread owns float4 (c,y,x4) AND (c,y+2048,x4) -- the JAX threefry
// counter pair (j, j+half). Unmasked tiles go through the CDNA5 async global<->LDS
// data mover (ASYNCcnt); masked tiles get threefry+erfinv noise computed while the
// async copies are in flight.
__global__ __launch_bounds__(256) void mae_apply_kernel(const float* __restrict__ img,
                                                        float* __restrict__ out,
                                                        const unsigned char* __restrict__ mask,
                                                        uint32_t kn0, uint32_t kn1) {
  __shared__ float4 smem[512];  // 2 x float4 per lane
  const int tid = threadIdx.x;
  const int t   = blockIdx.x * 256 + tid;   // 3 * 2048 * 1024 threads
  const int xq  = t & 1023;                 // x / 4
  const int y   = (t >> 10) & 2047;         // top-half row
  const int c   = t >> 21;                  // channel
  const int x4  = xq << 2;

  const int ptop = ((y >> 4) << 8) + (x4 >> 4);     // patch id (top), < 32768
  const int mt = mask[ptop];
  const int mb = mask[ptop + 32768];

  const size_t off_top = ((size_t)c << 24) + ((size_t)y << 12) + (size_t)x4;
  const size_t off_bot = off_top + ((size_t)2048 << 12);

  const uint32_t lds_top = (uint32_t)(uintptr_t)(&smem[tid * 2]);  // low 32b = LDS offset
  const uint32_t lds_bot = lds_top + 16u;

  // Stage 1: fire async DMA copies for unmasked tiles (VGPR-free, ASYNCcnt-tracked).
  if (!mt) {
    asm volatile("global_load_async_to_lds_b128 %0, %1, off"
                 :: "v"(lds_top), "v"(img + off_top) : "memory");
  }
  if (!mb) {
    asm volatile("global_load_async_to_lds_b128 %0, %1, off"
                 :: "v"(lds_bot), "v"(img + off_bot) : "memory");
  }

  // Stage 2: RNG compute for masked tiles overlaps the in-flight async loads.
  if (mt | mb) {
    // patchified element index k = py*48 + px*3 + c ; flat j = p*768 + k (< half)
    const uint32_t kbase = (uint32_t)((y & 15) * 48 + (x4 & 15) * 3 + c);
    const uint32_t jbase = (uint32_t)ptop * 768u + kbase;
    float tv0, tv1, tv2, tv3, bv0, bv1, bv2, bv3;
    uint32_t a, b;
    threefry2x32(kn0, kn1, jbase + 0u, jbase + 0u + 25165824u, a, b);
    tv0 = jax_normal(a); bv0 = jax_normal(b);
    threefry2x32(kn0, kn1, jbase + 3u, jbase + 3u + 25165824u, a, b);
    tv1 = jax_normal(a); bv1 = jax_normal(b);
    threefry2x32(kn0, kn1, jbase + 6u, jbase + 6u + 25165824u, a, b);
    tv2 = jax_normal(a); bv2 = jax_normal(b);
    threefry2x32(kn0, kn1, jbase + 9u, jbase + 9u + 25165824u, a, b);
    tv3 = jax_normal(a); bv3 = jax_normal(b);
    if (mt) *(float4*)(out + off_top) = make_float4(tv0, tv1, tv2, tv3);
    if (mb) *(float4*)(out + off_bot) = make_float4(bv0, bv1, bv2, bv3);
  }

  // Stage 3: drain async loads, then fire async stores (drained by s_endpgm's wait-idle).
  asm volatile("s_wait_asynccnt 0x0" ::: "memory");
  if (!mt) {
    asm volatile("global_store_async_from_lds_b128 %0, %1, off"
                 :: "v"(out + off_top), "v"(lds_top) : "memory");
  }
  if (!mb) {
    asm volatile("global_store_async_from_lds_b128 %0, %1, off"
                 :: "v"(out + off_bot), "v"(lds_bot) : "memory");
  }
}

extern "C" void kernel_launch(void* const* d_in, const int* in_sizes, int n_in,
                              void* d_out, int out_size, void* d_ws, size_t ws_size,
                              hipStream_t stream) {
  (void)in_sizes; (void)n_in; (void)out_size; (void)ws_size;
  const float* img = (const float*)d_in[0];
  float* out = (float*)d_out;
  unsigned char* mask = (unsigned char*)d_ws;  // 65536 bytes of scratch

  // Host-side (deterministic, capture-safe) JAX key derivation:
  // key(42) = (0,42); split -> threefry on counts [0,1,2,3] paired as (0,2),(1,3).
  uint32_t a0, b0, a1, b1;
  threefry2x32(0u, 42u, 0u, 2u, a0, b0);
  threefry2x32(0u, 42u, 1u, 3u, a1, b1);
  const uint32_t km0 = a0, km1 = a1;   // kmask = (out_x0 lane0, out_x0 lane1)
  const uint32_t kn0 = b0, kn1 = b1;   // knoise = (out_x1 lane0, out_x1 lane1)

  mae_mask_kernel<<<128, 256, 0, stream>>>(mask, km0, km1);
  mae_apply_kernel<<<24576, 256, 0, stream>>>(img, out, mask, kn0, kn1);
}